// MyGCN_27247272526306
// MI455X (gfx1250) — compile-verified
//
#include <hip/hip_runtime.h>
#include <math.h>

#define IN_F  128
#define H_F   256
#define OUT_F 40

typedef float v2f __attribute__((ext_vector_type(2)));
typedef float v8f __attribute__((ext_vector_type(8)));

__device__ __forceinline__ void atomAddF(float* p, float v) {
    unsafeAtomicAdd(p, v);   // hardware global_atomic_add_f32
}

// ---------------- degree / norm ----------------
__global__ void k_fill1(float* __restrict__ d, int n) {
    int i = blockIdx.x * blockDim.x + threadIdx.x;
    if (i < n) d[i] = 1.0f;               // self-loop contributes 1 to degree
}

__global__ void k_deg(const int* __restrict__ dst, float* __restrict__ deg, int E) {
    int e = blockIdx.x * blockDim.x + threadIdx.x;
    if (e < E) atomAddF(&deg[dst[e]], 1.0f);
}

__global__ void k_rsqrt(float* __restrict__ d, int n) {
    int i = blockIdx.x * blockDim.x + threadIdx.x;
    if (i < n) d[i] = rsqrtf(d[i]);       // deg >= 1 always (self loops)
}

// ---------------- WMMA GEMM: C[M,F] = A[M,K] @ B[K,F] ----------------
// Block = 128 threads (4 waves). Each wave owns one 16-row tile and TPW
// column tiles (tileN = wave + 4*t), reusing the A fragment across all of
// them each k-step. Requires M % 16 == 0 (true: N = 100000 = 6250*16).
template<int K, int F>
__global__ __launch_bounds__(128) void k_gemm(const float* __restrict__ A,
                                              const float* __restrict__ B,
                                              float* __restrict__ C, int M) {
    constexpr int colTiles = (F + 15) / 16;        // 16 for F=256, 3 for F=40
    constexpr int TPW      = (colTiles + 3) / 4;   // col tiles per wave

    const int lane = threadIdx.x & 31;
    const int wave = threadIdx.x >> 5;
    if (wave >= colTiles) return;                  // wave-uniform (F=40: wave 3 idles)

    const int half = lane >> 4;                    // 0: lanes 0-15, 1: lanes 16-31
    const int sub  = lane & 15;
    const int tileM = blockIdx.x;

    const float* arow = A + ((size_t)tileM * 16 + sub) * K;

    int  ncol[TPW];
    bool colOK[TPW];
    #pragma unroll
    for (int t = 0; t < TPW; ++t) {
        const int tileN = wave + 4 * t;            // strided so all waves stay busy
        ncol[t]  = tileN * 16 + sub;
        colOK[t] = (F % 16 == 0) || (ncol[t] < F); // constexpr-true for F=256
    }

    v8f acc[TPW];
    #pragma unroll
    for (int t = 0; t < TPW; ++t) acc[t] = (v8f){};

    for (int k0 = 0; k0 < K; k0 += 4) {
        const int ka = k0 + 2 * half;              // lanes 0-15: K=k0,k0+1; 16-31: +2,+3
        v2f a;
        a.x = arow[ka];
        a.y = arow[ka + 1];
        const float* b0 = B + (size_t)ka * F;
        const float* b1 = B + (size_t)(ka + 1) * F;
        #pragma unroll
        for (int t = 0; t < TPW; ++t) {
            v2f b;
            if (colOK[t]) { b.x = b0[ncol[t]]; b.y = b1[ncol[t]]; }
            else          { b.x = 0.0f;        b.y = 0.0f;        }
            acc[t] = __builtin_amdgcn_wmma_f32_16x16x4_f32(false, a, false, b,
                                                           (short)0, acc[t],
                                                           false, false);
        }
    }

    // C/D layout: VGPR j -> row tileM*16 + j + 8*half, col = ncol
    #pragma unroll
    for (int t = 0; t < TPW; ++t) {
        if (!colOK[t]) continue;                   // folds away for F=256
        float* crow = C + ((size_t)tileM * 16 + 8 * half) * F + ncol[t];
        #pragma unroll
        for (int j = 0; j < 8; ++j)
            crow[(size_t)j * F] = acc[t][j];
    }
}

// ---------------- self-loop init: agg = h * dinv^2 ----------------
__global__ void k_selfloop(const float* __restrict__ h, const float* __restrict__ dinv,
                           float* __restrict__ agg, long total, int F) {
    long idx = (long)blockIdx.x * blockDim.x + threadIdx.x;
    if (idx >= total) return;
    int i = (int)(idx / F);
    float d = dinv[i];
    agg[idx] = h[idx] * d * d;
}

// ---------------- edge scatter: agg[dst] += h[src] * norm ----------------
// One thread handles one float4 chunk of one edge's feature row.
__global__ void k_scatter(const int* __restrict__ src, const int* __restrict__ dst,
                          const float* __restrict__ dinv, const float* __restrict__ h,
                          float* __restrict__ agg, int E, int F, int F4) {
    long gid = (long)blockIdx.x * blockDim.x + threadIdx.x;
    long total = (long)E * F4;
    if (gid >= total) return;
    int e = (int)(gid / F4);
    int c = (int)(gid % F4) * 4;
    int s = src[e], d = dst[e];
    float nrm = dinv[s] * dinv[d];
    const float4 v = *reinterpret_cast<const float4*>(h + (size_t)s * F + c);
    float* outp = agg + (size_t)d * F + c;
    atomAddF(outp + 0, v.x * nrm);
    atomAddF(outp + 1, v.y * nrm);
    atomAddF(outp + 2, v.z * nrm);
    atomAddF(outp + 3, v.w * nrm);
}

// ---------------- bias + tanh (in place) ----------------
__global__ void k_bias_tanh(float* __restrict__ h, const float* __restrict__ b,
                            long total, int F) {
    long idx = (long)blockIdx.x * blockDim.x + threadIdx.x;
    if (idx >= total) return;
    int f = (int)(idx % F);
    h[idx] = tanhf(h[idx] + b[f]);
}

// ---------------- bias + log_softmax over 40 classes ----------------
__global__ void k_logsoftmax(const float* __restrict__ agg, const float* __restrict__ bias,
                             float* __restrict__ out, int n, int F) {
    int i = blockIdx.x * blockDim.x + threadIdx.x;
    if (i >= n) return;
    const float* row = agg + (size_t)i * F;
    float m = -INFINITY;
    for (int f = 0; f < F; ++f) m = fmaxf(m, row[f] + bias[f]);
    float s = 0.0f;
    for (int f = 0; f < F; ++f) s += expf(row[f] + bias[f] - m);
    float lse = m + logf(s);
    for (int f = 0; f < F; ++f) out[(size_t)i * F + f] = row[f] + bias[f] - lse;
}

extern "C" void kernel_launch(void* const* d_in, const int* in_sizes, int n_in,
                              void* d_out, int out_size, void* d_ws, size_t ws_size,
                              hipStream_t stream) {
    const float* x  = (const float*)d_in[0];
    const int*   ei = (const int*)  d_in[1];
    const float* W0 = (const float*)d_in[2];
    const float* b0 = (const float*)d_in[3];
    const float* W1 = (const float*)d_in[4];
    const float* b1 = (const float*)d_in[5];
    const float* W2 = (const float*)d_in[6];
    const float* b2 = (const float*)d_in[7];
    float* out = (float*)d_out;

    const int N = in_sizes[0] / IN_F;
    const int E = in_sizes[1] / 2;
    const int* src = ei;          // edge_index[0]
    const int* dst = ei + E;      // edge_index[1]

    float* ws   = (float*)d_ws;
    float* dinv = ws;                          // N
    float* bufA = ws + N;                      // N * H  (GEMM output)
    float* bufB = bufA + (size_t)N * H_F;      // N * H  (aggregation / activations)

    const int TB = 256;
    const long totH = (long)N * H_F;
    const long totO = (long)N * OUT_F;
    const long scH  = (long)E * (H_F / 4);
    const long scO  = (long)E * (OUT_F / 4);

    // ---- gcn_norm: deg -> dinv ----
    k_fill1 <<<(N + TB - 1) / TB, TB, 0, stream>>>(dinv, N);
    k_deg   <<<(E + TB - 1) / TB, TB, 0, stream>>>(dst, dinv, E);
    k_rsqrt <<<(N + TB - 1) / TB, TB, 0, stream>>>(dinv, N);

    dim3 blk(128);
    dim3 gM(N / 16, 1);   // one block (4 waves x TPW col tiles) covers full width

    // ---- layer 0 ----
    k_gemm<IN_F, H_F><<<gM, blk, 0, stream>>>(x, W0, bufA, N);
    k_selfloop <<<(totH + TB - 1) / TB, TB, 0, stream>>>(bufA, dinv, bufB, totH, H_F);
    k_scatter  <<<(scH + TB - 1) / TB, TB, 0, stream>>>(src, dst, dinv, bufA, bufB, E, H_F, H_F / 4);
    k_bias_tanh<<<(totH + TB - 1) / TB, TB, 0, stream>>>(bufB, b0, totH, H_F);

    // ---- layer 1 ----
    k_gemm<H_F, H_F><<<gM, blk, 0, stream>>>(bufB, W1, bufA, N);
    k_selfloop <<<(totH + TB - 1) / TB, TB, 0, stream>>>(bufA, dinv, bufB, totH, H_F);
    k_scatter  <<<(scH + TB - 1) / TB, TB, 0, stream>>>(src, dst, dinv, bufA, bufB, E, H_F, H_F / 4);
    k_bias_tanh<<<(totH + TB - 1) / TB, TB, 0, stream>>>(bufB, b1, totH, H_F);

    // ---- layer 2 + log_softmax ----
    k_gemm<H_F, OUT_F><<<gM, blk, 0, stream>>>(bufB, W2, bufA, N);
    k_selfloop <<<(totO + TB - 1) / TB, TB, 0, stream>>>(bufA, dinv, bufB, totO, OUT_F);
    k_scatter  <<<(scO + TB - 1) / TB, TB, 0, stream>>>(src, dst, dinv, bufA, bufB, E, OUT_F, OUT_F / 4);
    k_logsoftmax<<<(N + TB - 1) / TB, TB, 0, stream>>>(bufB, b2, out, N, OUT_F);

    (void)n_in; (void)out_size; (void)ws_size;
}